// MultiHeadSelfAttention_79345225826869
// MI455X (gfx1250) — compile-verified
//
#include <hip/hip_runtime.h>

typedef __attribute__((ext_vector_type(2))) float v2f;
typedef __attribute__((ext_vector_type(8))) float v8f;
typedef __attribute__((ext_vector_type(4))) unsigned int u32x4;
typedef __attribute__((ext_vector_type(8))) int i32x8;
typedef __attribute__((ext_vector_type(4))) int i32x4;

#define SEQ   2048
#define DM    768
#define NH    12
#define DH    64
#define BSZ   2
#define QKV_LD 2304   // 3*DM
#define PSTR  2052    // scores strip row stride (2048 + 4 pad)

#define KCH   64      // GEMM K-chunk staged per TDM transfer
#define BSTR  72      // LDS row stride for B tile (64 data + 8 pad dwords -> conflict-free)

#if __has_builtin(__builtin_amdgcn_tensor_load_to_lds) && \
    __has_builtin(__builtin_amdgcn_s_wait_tensorcnt)
#define HAVE_TDM 1
#else
#define HAVE_TDM 0
#endif

// D = A(16x4) * B(4x16) + C, fp32, wave32.
// A frag (2 VGPR): lane<16: row=lane, {K=k0,k0+1}; lane>=16: row=lane-16, {K=k0+2,k0+3}
// B frag (2 VGPR): lane<16: col=lane, {K=k0,k0+1}; lane>=16: col=lane-16, {K=k0+2,k0+3}
// C/D (8 VGPR): vgpr j -> M = j + 8*(lane>=16), N = lane&15
__device__ __forceinline__ v8f wmma4(v2f a, v2f b, v8f c) {
  return __builtin_amdgcn_wmma_f32_16x16x4_f32(
      false, a, false, b, (short)0, c, false, false);
}

#if HAVE_TDM
// TDM: async 2D f32 tile (tile_x x tile_y) global -> LDS, row stride
// stride_elems in memory, LDS rows padded 64->72 dwords via pad fields.
// D# bitfields per CDNA5 ISA ch.8 (group0: count/lds/global/type, group1: dims).
// This toolchain exposes the 6-arg builtin: (g0,u32x4)(g1,i32x8)(g2,i32x4)(g3,i32x4)(g4,i32x8)(cpol,i32)
__device__ __forceinline__ void tdm_load_tile_f32(
    unsigned lds_byte_off, const float* gptr,
    unsigned tile_x, unsigned tile_y, unsigned long long stride_elems) {
  unsigned long long ga = (unsigned long long)(uintptr_t)gptr;
  u32x4 g0;
  g0[0] = 1u;                                         // count=1, user D#
  g0[1] = lds_byte_off;                               // lds_addr (bytes)
  g0[2] = (unsigned)(ga & 0xffffffffull);             // global_addr[31:0]
  g0[3] = (unsigned)((ga >> 32) & 0x1ffffffull)       // global_addr[56:32]
        | (2u << 30);                                 // type=2 ("image")
  // group1 word0: data_size=2 (4B) | pad_enable | pad_interval=64DW (code 5)
  //               | pad_amount=8DW (code 7)
  unsigned w0 = (2u << 16) | (1u << 20) | (5u << 22) | (7u << 25);
  i32x8 g1;
  g1[0] = (int)w0;
  g1[1] = (int)((tile_x & 0xffffu) << 16);            // tensor_dim0[15:0]
  g1[2] = (int)(((tile_x >> 16) & 0xffffu) | ((tile_y & 0xffffu) << 16)); // dim0 hi | dim1 lo
  g1[3] = (int)(((tile_y >> 16) & 0xffffu) | ((tile_x & 0xffffu) << 16)); // dim1 hi | tile_dim0
  g1[4] = (int)(tile_y & 0xffffu);                    // tile_dim1 (tile_dim2=0)
  g1[5] = (int)(stride_elems & 0xffffffffull);        // tensor_dim0_stride lo
  g1[6] = (int)((stride_elems >> 32) & 0xffffull);    // tensor_dim0_stride hi
  g1[7] = 0;
  i32x4 g2 = {0, 0, 0, 0};
  i32x4 g3 = {0, 0, 0, 0};
  i32x8 g4 = {0, 0, 0, 0, 0, 0, 0, 0};
  __builtin_amdgcn_tensor_load_to_lds(g0, g1, g2, g3, g4, 0);
}
__device__ __forceinline__ unsigned lds_off32(const void* p) {
  return (unsigned)((unsigned long long)(uintptr_t)p & 0xffffffffull); // LDS aperture: addr[31:0]
}
#endif

// C[MxN] = A[MxK] @ B[KxN] + bias[N], row-major. One wave per 16x64 tile.
// B tiles streamed to LDS by the Tensor Data Mover, double-buffered.
// grid: (M/16, N/64), block: 32 threads (1 wave). K multiple of KCH.
__global__ void gemm_bias(const float* __restrict__ A, const float* __restrict__ B,
                          const float* __restrict__ bias, float* __restrict__ C,
                          int M, int K, int N) {
  const int lane = threadIdx.x & 31;
  const int half = lane >> 4;
  const int lid  = lane & 15;
  const int m0 = blockIdx.x * 16;
  const int n0 = blockIdx.y * 64;

  v8f acc[4] = {};
  const float* arow = A + (size_t)(m0 + lid) * K;
  const int nch = K / KCH;

#if HAVE_TDM
  __shared__ float Blds[2][KCH * BSTR];
  // prologue: stream chunk 0
  tdm_load_tile_f32(lds_off32(&Blds[0][0]), B + (size_t)0 * N + n0, 64, KCH,
                    (unsigned long long)N);
  for (int c = 0; c < nch; ++c) {
    if (c + 1 < nch)
      tdm_load_tile_f32(lds_off32(&Blds[(c + 1) & 1][0]),
                        B + (size_t)(c + 1) * KCH * N + n0, 64, KCH,
                        (unsigned long long)N);
    if (c + 1 < nch)
      __builtin_amdgcn_s_wait_tensorcnt(1);   // oldest (chunk c) complete
    else
      __builtin_amdgcn_s_wait_tensorcnt(0);
    const float* Bl = &Blds[c & 1][0];
    const float* ar = arow + c * KCH;
    // hoist A fragments for the whole chunk: one load clause, overlaps DMA
    v2f af[KCH / 4];
#pragma unroll
    for (int i = 0; i < KCH / 4; ++i)
      af[i] = *(const v2f*)(ar + 4 * i + 2 * half);
#pragma unroll
    for (int i = 0; i < KCH / 4; ++i) {
      const int r = 4 * i + 2 * half;
#pragma unroll
      for (int t = 0; t < 4; ++t) {
        v2f bf;
        bf.x = Bl[(r    ) * BSTR + t * 16 + lid];
        bf.y = Bl[(r + 1) * BSTR + t * 16 + lid];
        acc[t] = wmma4(af[i], bf, acc[t]);
      }
    }
  }
#else
  for (int k0 = 0; k0 < K; k0 += 4) {
    v2f a = *(const v2f*)(arow + k0 + 2 * half);
    const float* b0 = B + (size_t)(k0 + 2 * half) * N;
    const float* b1 = b0 + N;
#pragma unroll
    for (int t = 0; t < 4; ++t) {
      int n = n0 + t * 16 + lid;
      v2f bf;
      bf.x = b0[n];
      bf.y = b1[n];
      acc[t] = wmma4(a, bf, acc[t]);
    }
  }
#endif

#pragma unroll
  for (int t = 0; t < 4; ++t) {
    int n = n0 + t * 16 + lid;
    float bv = bias[n];
#pragma unroll
    for (int j = 0; j < 8; ++j) {
      C[(size_t)(m0 + j + 8 * half) * N + n] = acc[t][j] + bv;
    }
  }
}

// Fused attention: one block per (16-query strip, head, batch).
// 512 threads = 16 waves. Dynamic LDS: P[16][PSTR] scores strip + red[16*256].
// prob is written to HBM exactly once; P@V runs out of LDS.
__global__ void mhsa_attn(const float* __restrict__ qkv, float* __restrict__ prob,
                          float* __restrict__ ctx) {
  extern __shared__ float smem[];
  float* P   = smem;                 // 16 * PSTR floats (~131 KB)
  float* red = smem + 16 * PSTR;     // 16 * 256 floats  (16 KB)

  const int tid  = threadIdx.x;
  const int w    = tid >> 5;
  const int lane = tid & 31;
  const int half = lane >> 4;
  const int lid  = lane & 15;

  const int q0 = blockIdx.x * 16;
  const int h  = blockIdx.y;
  const int b  = blockIdx.z;

  const size_t rowBase = (size_t)b * SEQ;
  const int qc = h * DH;
  const int kc = DM + h * DH;
  const int vc = 2 * DM + h * DH;
  const float scale = 0.125f;        // 1/sqrt(64)

  // ---------------- scores: wave w covers key columns [w*128, w*128+128) ----
  v2f afr[16];
  const float* qrow = qkv + (rowBase + q0 + lid) * QKV_LD + qc;
#pragma unroll
  for (int k = 0; k < 16; ++k)
    afr[k] = *(const v2f*)(qrow + 4 * k + 2 * half);

  for (int sub = 0; sub < 8; ++sub) {
    const int n0 = w * 128 + sub * 16;
    const float* krow = qkv + (rowBase + n0 + lid) * QKV_LD + kc;
    v8f acc = {};
#pragma unroll
    for (int k = 0; k < 16; ++k) {
      v2f bf = *(const v2f*)(krow + 4 * k + 2 * half);
      acc = wmma4(afr[k], bf, acc);
    }
#pragma unroll
    for (int j = 0; j < 8; ++j)
      P[(j + 8 * half) * PSTR + n0 + lid] = acc[j] * scale;
  }
  __syncthreads();

  // ---------------- softmax: wave w owns row w; single prob write ----------
  {
    float* row = P + w * PSTR;
    float m = -3.402823466e38f;
    for (int c = lane; c < SEQ; c += 32) m = fmaxf(m, row[c]);
#pragma unroll
    for (int o = 16; o > 0; o >>= 1) m = fmaxf(m, __shfl_xor(m, o, 32));
    float s = 0.f;
    for (int c = lane; c < SEQ; c += 32) {
      float e = __expf(row[c] - m);
      row[c] = e;
      s += e;
    }
#pragma unroll
    for (int o = 16; o > 0; o >>= 1) s += __shfl_xor(s, o, 32);
    const float inv = 1.f / s;
    float* pout = prob + (((size_t)(b * NH + h)) * SEQ + q0 + w) * SEQ;
    for (int c = lane; c < SEQ; c += 32) {
      float v = row[c] * inv;
      row[c] = v;
      pout[c] = v;
    }
  }
  __syncthreads();

  // ---------------- ctx = P @ V : wave w -> (ntile = w&3, kslice = w>>2) ----
  const int ntile  = w & 3;
  const int kslice = w >> 2;
  const int vcol = vc + ntile * 16 + lid;
  v8f cacc = {};
  const int kbeg = kslice * 512;
  for (int k0 = kbeg; k0 < kbeg + 512; k0 += 4) {
    v2f a = *(const v2f*)(P + lid * PSTR + k0 + 2 * half);
    const int kk = k0 + 2 * half;
    v2f bf;
    bf.x = qkv[(rowBase + kk) * QKV_LD + vcol];
    bf.y = qkv[(rowBase + kk + 1) * QKV_LD + vcol];
    cacc = wmma4(a, bf, cacc);
  }
#pragma unroll
  for (int j = 0; j < 8; ++j)
    red[w * 256 + j * 32 + lane] = cacc[j];
  __syncthreads();

  if (w < 4) {
#pragma unroll
    for (int j = 0; j < 8; ++j) {
      float sum = red[(w     ) * 256 + j * 32 + lane]
                + red[(w +  4) * 256 + j * 32 + lane]
                + red[(w +  8) * 256 + j * 32 + lane]
                + red[(w + 12) * 256 + j * 32 + lane];
      ctx[((size_t)(b * SEQ) + q0 + j + 8 * half) * DM + h * DH + w * 16 + lid] = sum;
    }
  }
}

extern "C" void kernel_launch(void* const* d_in, const int* in_sizes, int n_in,
                              void* d_out, int out_size, void* d_ws, size_t ws_size,
                              hipStream_t stream) {
  const float* inp  = (const float*)d_in[0];  // [2,2048,768]
  const float* Wqkv = (const float*)d_in[1];  // [768,2304]
  const float* bqkv = (const float*)d_in[2];  // [2304]
  const float* Wout = (const float*)d_in[3];  // [768,768]
  const float* bout = (const float*)d_in[4];  // [768]

  float* prob = (float*)d_out;                                  // [24,2048,2048]
  float* out  = prob + (size_t)BSZ * NH * SEQ * SEQ;            // [2,2048,768]

  float* qkv = (float*)d_ws;                                    // [4096,2304]
  float* ctx = qkv + (size_t)BSZ * SEQ * QKV_LD;                // [4096,768]

  // 1) qkv = inp @ W_qkv + b_qkv
  {
    dim3 grid((BSZ * SEQ) / 16, (3 * DM) / 64);
    gemm_bias<<<grid, 32, 0, stream>>>(inp, Wqkv, bqkv, qkv, BSZ * SEQ, DM, 3 * DM);
  }
  // 2) fused scores -> softmax -> prob write + ctx
  {
    dim3 grid(SEQ / 16, NH, BSZ);
    size_t shmem = (size_t)(16 * PSTR + 16 * 256) * sizeof(float);
    mhsa_attn<<<grid, 512, shmem, stream>>>(qkv, prob, ctx);
  }
  // 3) out = ctx @ W_out + b_out
  {
    dim3 grid((BSZ * SEQ) / 16, DM / 64);
    gemm_bias<<<grid, 32, 0, stream>>>(ctx, Wout, bout, out, BSZ * SEQ, DM, DM);
  }
}